// Attention_24266565222906
// MI455X (gfx1250) — compile-verified
//
#include <hip/hip_runtime.h>
#include <math.h>

typedef float v2f __attribute__((ext_vector_type(2)));
typedef float v8f __attribute__((ext_vector_type(8)));

#define BATCH 32
#define SEQ   2048
#define DIM   1024
#define EPS_K 1e-7f

// ---------------------------------------------------------------------------
// Kernel 1: proj[b,e] = sum_d y[b,d] * M[e,d]   (C = Y * M^T, 32x1024x1024)
// One wave32 per 16x16 output tile using V_WMMA_F32_16X16X4_F32.
// ---------------------------------------------------------------------------
__global__ __launch_bounds__(128) void proj_wmma_kernel(
    const float* __restrict__ y, const float* __restrict__ Mm,
    float* __restrict__ proj) {
  const int wave = threadIdx.x >> 5;
  const int lane = threadIdx.x & 31;
  const int tile = blockIdx.x * 4 + wave;     // 128 tiles = 2 b-tiles x 64 e-tiles
  const int b0 = (tile >> 6) << 4;
  const int e0 = (tile & 63) << 4;
  const int half = lane >> 4;                 // 0: lanes 0-15, 1: lanes 16-31
  const int lh = lane & 15;

  // A (16x4 f32): vgpr j, low lanes = A[m=lh, k=k0+j], high lanes = k0+2+j
  // B (4x16 f32): vgpr j, low lanes = B[k=k0+j, n=lh], high lanes = k0+2+j
  // -> both fragments load as float2 at row*D + k0 + 2*half
  const float* yrow = y + (size_t)(b0 + lh) * DIM + 2 * half;
  const float* mrow = Mm + (size_t)(e0 + lh) * DIM + 2 * half;

  v8f c = {0.f, 0.f, 0.f, 0.f, 0.f, 0.f, 0.f, 0.f};
  for (int k0 = 0; k0 < DIM; k0 += 4) {
    float2 av = *(const float2*)(yrow + k0);
    float2 bv = *(const float2*)(mrow + k0);
    v2f a; a.x = av.x; a.y = av.y;
    v2f b; b.x = bv.x; b.y = bv.y;
    c = __builtin_amdgcn_wmma_f32_16x16x4_f32(false, a, false, b, (short)0, c,
                                              false, false);
  }
  // D (16x16 f32): vgpr r, low lanes = D[m=r, n=lh], high lanes = D[m=r+8, n=lh]
#pragma unroll
  for (int r = 0; r < 8; ++r) {
    proj[(size_t)(b0 + r + 8 * half) * DIM + e0 + lh] = c[r];
  }
}

// ---------------------------------------------------------------------------
// Kernel 2: e[b,s] = exp(tanh(x[b,s,:] . proj[b,:])) * mask[b,s]
// One wave per row; float4 lane-strided loads (512B contiguous per issue).
// Per-block partial sums -> partial[b*16 + s_tile].
// ---------------------------------------------------------------------------
__global__ __launch_bounds__(256) void eij_kernel(
    const float* __restrict__ x, const float* __restrict__ proj,
    const int* __restrict__ mask, float* __restrict__ e_out,
    float* __restrict__ partial) {
  const int b = blockIdx.y;
  const int s_base = blockIdx.x * 128;
  const int wave = threadIdx.x >> 5;
  const int lane = threadIdx.x & 31;

  const float4* projv = (const float4*)(proj + (size_t)b * DIM);
  float wsum = 0.f;

  for (int r = 0; r < 16; ++r) {
    const int s = s_base + wave * 16 + r;
    const float* xrow = x + ((size_t)b * SEQ + s) * DIM;
    const float4* xv = (const float4*)xrow;
    if (r + 1 < 16) {  // prefetch next row (global_prefetch_b8)
      __builtin_prefetch(xrow + DIM + lane * 32, 0, 0);
    }
    float4 acc = make_float4(0.f, 0.f, 0.f, 0.f);
#pragma unroll
    for (int i = 0; i < 8; ++i) {
      float4 xx = xv[lane + 32 * i];
      float4 pp = projv[lane + 32 * i];
      acc.x += xx.x * pp.x;
      acc.y += xx.y * pp.y;
      acc.z += xx.z * pp.z;
      acc.w += xx.w * pp.w;
    }
    float d = (acc.x + acc.y) + (acc.z + acc.w);
#pragma unroll
    for (int off = 16; off > 0; off >>= 1) d += __shfl_xor(d, off, 32);
    float ev = expf(tanhf(d)) * (float)mask[b * SEQ + s];
    wsum += ev;                       // same value in all lanes
    if (lane == 0) e_out[b * SEQ + s] = ev;
  }

  __shared__ float red[8];
  if (lane == 0) red[wave] = wsum;
  __syncthreads();
  if (threadIdx.x == 0) {
    float t = 0.f;
#pragma unroll
    for (int w = 0; w < 8; ++w) t += red[w];
    partial[b * 16 + blockIdx.x] = t;
  }
}

// ---------------------------------------------------------------------------
// Kernel 3: sums[b] = sum of 16 partials (deterministic, no float atomics)
// ---------------------------------------------------------------------------
__global__ void sum_kernel(const float* __restrict__ partial,
                           float* __restrict__ sums) {
  const int b = threadIdx.x;
  if (b < BATCH) {
    float t = 0.f;
#pragma unroll
    for (int i = 0; i < 16; ++i) t += partial[b * 16 + i];
    sums[b] = t;
  }
}

// ---------------------------------------------------------------------------
// Kernel 4: a[b,s] = e[b,s] / (sums[b] + EPS), in place on d_out
// ---------------------------------------------------------------------------
__global__ __launch_bounds__(256) void norm_kernel(float* __restrict__ out,
                                                   const float* __restrict__ sums) {
  const int idx = blockIdx.x * 256 + threadIdx.x;
  const int b = idx >> 11;  // / SEQ
  out[idx] = out[idx] / (sums[b] + EPS_K);
}

extern "C" void kernel_launch(void* const* d_in, const int* in_sizes, int n_in,
                              void* d_out, int out_size, void* d_ws,
                              size_t ws_size, hipStream_t stream) {
  const float* x = (const float*)d_in[0];     // [32,2048,1024] f32
  const float* y = (const float*)d_in[1];     // [32,1024] f32
  const int* mask = (const int*)d_in[2];      // [32,2048] i32
  const float* Mm = (const float*)d_in[3];    // [1024,1024] f32
  float* out = (float*)d_out;                 // [32,2048] f32

  float* proj = (float*)d_ws;                 // 32*1024 floats
  float* partial = proj + BATCH * DIM;        // 32*16 floats
  float* sums = partial + BATCH * 16;         // 32 floats

  proj_wmma_kernel<<<32, 128, 0, stream>>>(y, Mm, proj);
  eij_kernel<<<dim3(16, BATCH), 256, 0, stream>>>(x, proj, mask, out, partial);
  sum_kernel<<<1, 32, 0, stream>>>(partial, sums);
  norm_kernel<<<(BATCH * SEQ) / 256, 256, 0, stream>>>(out, sums);
}